// SpatialDisplConv_12309376270993
// MI455X (gfx1250) — compile-verified
//
#include <hip/hip_runtime.h>

#ifndef __has_builtin
#define __has_builtin(x) 0
#endif

#define AS1 __attribute__((address_space(1)))
#define AS3 __attribute__((address_space(3)))

typedef AS1 int* global_i32_ptr;
typedef AS3 int* lds_i32_ptr;

// ---------------- problem constants (from reference) ----------------
constexpr int BATCH = 4;
constexpr int CH    = 3;
constexpr int H     = 512;
constexpr int W     = 512;
constexpr int K     = 11;          // KSIZE
constexpr int Hp    = H + K - 1;   // 522
constexpr int Wp    = W + K - 1;   // 522

// ---------------- tiling ----------------
constexpr int TX    = 32;          // output tile width
constexpr int TY    = 32;          // output tile height
constexpr int HALO  = 6;           // covers |displacement| < 6 sigma; slow path handles rest
constexpr int SH    = TY + K + 2 * HALO;   // 55 staged rows
constexpr int SW    = TX + K + 2 * HALO;   // 55 staged cols
constexpr int SWP   = SW + 1;              // 56 floats: row stride = 224B (16B multiple)
constexpr int NTHREADS = 256;

// ---------------- CDNA5 async global->LDS staging ----------------
__device__ __forceinline__ void stage_async_b32(const float* g, float* l) {
#if __has_builtin(__builtin_amdgcn_global_load_async_to_lds_b32)
  __builtin_amdgcn_global_load_async_to_lds_b32((global_i32_ptr)(g),
                                                (lds_i32_ptr)(l),
                                                /*imm offset=*/0, /*cpol=*/0);
#else
  *l = *g;   // fallback: plain copy through VGPRs
#endif
}

__device__ __forceinline__ void wait_async_all() {
#if __has_builtin(__builtin_amdgcn_global_load_async_to_lds_b32)
#  if __has_builtin(__builtin_amdgcn_s_wait_asynccnt)
  __builtin_amdgcn_s_wait_asynccnt(0);
#  else
  asm volatile("s_wait_asynccnt 0" ::: "memory");
#  endif
#endif
}

__global__ __launch_bounds__(NTHREADS) void sdc_kernel(
    const float* __restrict__ img,   // input1 (B,C,Hp,Wp)
    const float* __restrict__ kv,    // input2 (B,K,H,W)  vertical taps
    const float* __restrict__ kh,    // input3 (B,K,H,W)  horizontal taps
    const float* __restrict__ disp,  // input4 (B,2,H,W)  [0]=dx, [1]=dy
    float* __restrict__ out)         // (B,C,H,W)
{
  __shared__ __align__(16) float tile[CH][SH][SWP];

  const int b    = blockIdx.z;
  const int x0   = blockIdx.x * TX;
  const int y0   = blockIdx.y * TY;
  const int regY = y0 - HALO;
  const int regX = x0 - HALO;
  const int tid  = threadIdx.x;

  const float* imgb = img + (unsigned)b * (CH * Hp * Wp);

  // ---- stage input1 region (3 channels, SHxSW, image-bounds masked) ----
  const bool crosses = (regY < 0) | (regX < 0) | (regY + SH > Hp) | (regX + SW > Wp);
  if (crosses) {
    for (int i = tid; i < CH * SH * SWP; i += NTHREADS)
      ((float*)tile)[i] = 0.0f;
    __syncthreads();   // zeros visible before async loads overwrite valid lanes
  }
  // pad column (col SW) must always be zero: it is read by b128 gathers with tap weight 0
  for (int i = tid; i < CH * SH; i += NTHREADS)
    tile[i / SH][i % SH][SW] = 0.0f;

  for (int i = tid; i < CH * SH * SW; i += NTHREADS) {
    const int c   = i / (SH * SW);
    const int r   = (i / SW) % SH;
    const int col = i % SW;
    const int gy  = regY + r;
    const int gx  = regX + col;
    if ((unsigned)gy < (unsigned)Hp && (unsigned)gx < (unsigned)Wp) {
      stage_async_b32(imgb + (unsigned)(c * (Hp * Wp) + gy * Wp + gx),
                      &tile[c][r][col]);
    }
  }
  wait_async_all();    // ASYNCcnt==0: this wave's LDS writes have landed
  __syncthreads();     // all waves' staging (and pad zeros) visible

  // ---- per-pixel compute: 4 pixels per thread ----
  const int txl = tid & (TX - 1);   // 0..31
  const int ty0 = tid >> 5;         // 0..7
  constexpr unsigned plane = (unsigned)H * W;

  for (int py = ty0; py < TY; py += NTHREADS / TX) {
    const int y = y0 + py;
    const int x = x0 + txl;
    const unsigned pix = (unsigned)y * W + (unsigned)x;

    const float dxv = disp[(unsigned)b * 2u * plane + pix];
    const float dyv = disp[((unsigned)b * 2u + 1u) * plane + pix];
    const float pyf = (float)y + dyv;
    const float pxf = (float)x + dxv;
    const float fy0 = floorf(pyf);
    const float fx0 = floorf(pxf);
    const float wy  = pyf - fy0;
    const float wx  = pxf - fx0;
    const int   iy0 = (int)fy0;
    const int   ix0 = (int)fx0;

    // interpolated 12-tap kernels (registers)
    float Vy[K + 1], Hx[K + 1];
    {
      const float* p2 = kv + (unsigned)b * (K * plane) + pix;
      const float* p3 = kh + (unsigned)b * (K * plane) + pix;
      float prev2 = 0.0f, prev3 = 0.0f;
      const float omwy = 1.0f - wy, omwx = 1.0f - wx;
#pragma unroll
      for (int g = 0; g < K; ++g) {
        const float v2 = p2[(unsigned)g * plane];
        const float v3 = p3[(unsigned)g * plane];
        Vy[g] = v2 * omwy + prev2 * wy;
        Hx[g] = v3 * omwx + prev3 * wx;
        prev2 = v2;
        prev3 = v3;
      }
      Vy[K] = prev2 * wy;
      Hx[K] = prev3 * wx;
    }

    float acc0 = 0.0f, acc1 = 0.0f, acc2 = 0.0f;
    const int ry0 = iy0 - regY;
    const int rx0 = ix0 - regX;

    if (ry0 >= 0 && rx0 >= 0 && (ry0 + K) < SH && (rx0 + K) < SW) {
      // ---- fast path: 12x12 window from LDS via aligned b128 reads ----
      const int off  = rx0 & 3;        // 0..3
      const int rx0a = rx0 - off;      // 16B-aligned column

      // Build 16-wide zero-padded taps: Hx16[off+g] = Hx[g], else 0.
      // Two cndmask funnel-shift stages (constant indices -> stays in VGPRs).
      float Hx16[16];
#pragma unroll
      for (int j = 0; j < 16; ++j) Hx16[j] = (j <= K) ? Hx[j] : 0.0f;
      const bool s1 = (off & 1) != 0;
      const bool s2 = (off & 2) != 0;
#pragma unroll
      for (int j = 15; j >= 1; --j) Hx16[j] = s1 ? Hx16[j - 1] : Hx16[j];
      Hx16[0] = s1 ? 0.0f : Hx16[0];
#pragma unroll
      for (int j = 15; j >= 2; --j) Hx16[j] = s2 ? Hx16[j - 2] : Hx16[j];
      Hx16[1] = s2 ? 0.0f : Hx16[1];
      Hx16[0] = s2 ? 0.0f : Hx16[0];

      for (int gy = 0; gy <= K; ++gy) {
        const float vy = Vy[gy];
        const int   ry = ry0 + gy;
        const float4* q0 = reinterpret_cast<const float4*>(&tile[0][ry][rx0a]);
        const float4* q1 = reinterpret_cast<const float4*>(&tile[1][ry][rx0a]);
        const float4* q2 = reinterpret_cast<const float4*>(&tile[2][ry][rx0a]);
        float rs0 = 0.0f, rs1 = 0.0f, rs2 = 0.0f;
#pragma unroll
        for (int qq = 0; qq < 4; ++qq) {
          const float4 a = q0[qq];
          const float4 bv = q1[qq];
          const float4 cv = q2[qq];
          const float h0 = Hx16[4 * qq + 0];
          const float h1 = Hx16[4 * qq + 1];
          const float h2 = Hx16[4 * qq + 2];
          const float h3 = Hx16[4 * qq + 3];
          rs0 += h0 * a.x + h1 * a.y + h2 * a.z + h3 * a.w;
          rs1 += h0 * bv.x + h1 * bv.y + h2 * bv.z + h3 * bv.w;
          rs2 += h0 * cv.x + h1 * cv.y + h2 * cv.z + h3 * cv.w;
        }
        acc0 += vy * rs0;
        acc1 += vy * rs1;
        acc2 += vy * rs2;
      }
    } else {
      // ---- slow path (rare: |displacement| >= HALO): bounds-checked global gather ----
      for (int gy = 0; gy <= K; ++gy) {
        const int  iy  = iy0 + gy;
        const bool okY = (unsigned)iy < (unsigned)Hp;
        const float vy = Vy[gy];
        const unsigned rowoff = (unsigned)(okY ? iy : 0) * Wp;
        float rs[CH] = {0.0f, 0.0f, 0.0f};
        for (int gx = 0; gx <= K; ++gx) {
          const int  ix = ix0 + gx;
          const bool ok = okY && ((unsigned)ix < (unsigned)Wp);
          const float hx = Hx[gx];
#pragma unroll
          for (int c = 0; c < CH; ++c) {
            const float v = ok ? imgb[(unsigned)c * (Hp * Wp) + rowoff + (unsigned)ix]
                               : 0.0f;
            rs[c] += hx * v;
          }
        }
        acc0 += vy * rs[0];
        acc1 += vy * rs[1];
        acc2 += vy * rs[2];
      }
    }

    out[((unsigned)b * CH + 0) * plane + pix] = acc0;
    out[((unsigned)b * CH + 1) * plane + pix] = acc1;
    out[((unsigned)b * CH + 2) * plane + pix] = acc2;
  }
}

extern "C" void kernel_launch(void* const* d_in, const int* in_sizes, int n_in,
                              void* d_out, int out_size, void* d_ws, size_t ws_size,
                              hipStream_t stream) {
  (void)in_sizes; (void)n_in; (void)out_size; (void)d_ws; (void)ws_size;
  const float* input1 = (const float*)d_in[0];
  const float* input2 = (const float*)d_in[1];
  const float* input3 = (const float*)d_in[2];
  const float* input4 = (const float*)d_in[3];
  float* outp = (float*)d_out;

  dim3 grid(W / TX, H / TY, BATCH);
  dim3 block(NTHREADS);
  sdc_kernel<<<grid, block, 0, stream>>>(input1, input2, input3, input4, outp);
}